// MyCrossEntropyLoss_9405978378353
// MI455X (gfx1250) — compile-verified
//
#include <hip/hip_runtime.h>

// Problem constants from the reference.
#define kN 32768
#define kC 4096

typedef float v2f __attribute__((ext_vector_type(2)));
typedef float v8f __attribute__((ext_vector_type(8)));

// Block = 256 threads = 8 waves. Each wave handles 16 rows x 2048 columns
// (half the row); wave pairs (2w, 2w+1) cover the two column halves of the
// same 16-row tile. Block therefore covers 64 rows -> grid = N/64 = 512.
//
// Per wave: A-matrix of V_WMMA_F32_16X16X4_F32 holds 16 rows x 4 K-elements
// of v = factor*exp(logits); B = ones, so the accumulator C collects the
// 16 row-sums (replicated across all 16 N-columns). Lane<16 supplies K,K+1,
// lane>=16 supplies K+2,K+3 for row M = lane&15 (per CDNA5 ISA A layout).
__global__ __launch_bounds__(256) void
weighted_lse_rowsum(const float* __restrict__ logits,
                    const float* __restrict__ factor,
                    const int* __restrict__ target,
                    float* __restrict__ block_partials)
{
    __shared__ float s_rows[8][16];  // [wave][row-in-tile] half-row sums
    __shared__ float s_red[64];

    const int tid  = threadIdx.x;
    const int wave = tid >> 5;
    const int lane = tid & 31;
    const int rowTile = wave >> 1;       // 0..3
    const int kHalf   = wave & 1;        // which 2048-column half
    const int m       = lane & 15;       // row within 16-row tile
    const int kOff    = (lane >> 4) * 2; // 0 (lanes 0-15) or 2 (lanes 16-31)

    const int rowBase = blockIdx.x * 64 + rowTile * 16;
    const int row     = rowBase + m;

    // Base in float2 units; all offsets are even so this is 8B aligned.
    const size_t elt = (size_t)row * kC + (size_t)kHalf * 2048 + kOff;
    const float2* __restrict__ lp2 = (const float2*)logits + (elt >> 1);
    const float2* __restrict__ fp2 = (const float2*)factor + (elt >> 1);

    v8f acc0 = {};
    v8f acc1 = {};
    v2f ones; ones.x = 1.0f; ones.y = 1.0f;

    // Per lane: 512 K-chunks of 4 columns; each chunk = one float2 per input,
    // chunk-to-chunk stride = 2 float2 (4 floats). Two accumulators for ILP.
#pragma unroll 4
    for (int k2 = 0; k2 < 1024; k2 += 4) {
        float2 l0 = lp2[k2];
        float2 f0 = fp2[k2];
        float2 l1 = lp2[k2 + 2];
        float2 f1 = fp2[k2 + 2];
        v2f a0, a1;
        a0.x = f0.x * __expf(l0.x);
        a0.y = f0.y * __expf(l0.y);
        a1.x = f1.x * __expf(l1.x);
        a1.y = f1.y * __expf(l1.y);
        acc0 = __builtin_amdgcn_wmma_f32_16x16x4_f32(false, a0, false, ones,
                                                     (short)0, acc0, false, false);
        acc1 = __builtin_amdgcn_wmma_f32_16x16x4_f32(false, a1, false, ones,
                                                     (short)0, acc1, false, false);
    }

    v8f accT = acc0 + acc1;

    // C layout (16x16 f32): lane n (0-15) VGPR r -> (M=r, N=n);
    // lane n+16 VGPR r -> (M=r+8, N=n). Column N=0 lives in lanes 0 and 16.
    if (lane == 0) {
#pragma unroll
        for (int j = 0; j < 8; ++j) s_rows[wave][j] = accT[j];
    } else if (lane == 16) {
#pragma unroll
        for (int j = 0; j < 8; ++j) s_rows[wave][8 + j] = accT[j];
    }
    __syncthreads();

    // Threads 0..63: combine the two column-half sums per row, then
    // c = log(S) - logits[row, target[row]].
    if (tid < 64) {
        const int tile = tid >> 4;
        const int mr   = tid & 15;
        const float S  = s_rows[tile * 2 + 0][mr] + s_rows[tile * 2 + 1][mr];
        const int r    = blockIdx.x * 64 + tile * 16 + mr;
        const int tgt  = target[r];
        const float picked = logits[(size_t)r * kC + tgt];
        s_red[tid] = logf(S) - picked;
    }
    __syncthreads();

    // Deterministic fixed-order tree reduction of the 64 row contributions.
    for (int s = 32; s > 0; s >>= 1) {
        if (tid < s) s_red[tid] += s_red[tid + s];
        __syncthreads();
    }
    if (tid == 0) block_partials[blockIdx.x] = s_red[0];
}

// Reduce the 512 block partials deterministically and scale by 1/N.
__global__ __launch_bounds__(512) void
weighted_lse_finalize(const float* __restrict__ partials, float* __restrict__ out)
{
    __shared__ float red[512];
    const int t = threadIdx.x;
    red[t] = partials[t];
    __syncthreads();
    for (int s = 256; s > 0; s >>= 1) {
        if (t < s) red[t] += red[t + s];
        __syncthreads();
    }
    if (t == 0) out[0] = red[0] * (1.0f / (float)kN);
}

extern "C" void kernel_launch(void* const* d_in, const int* in_sizes, int n_in,
                              void* d_out, int out_size, void* d_ws, size_t ws_size,
                              hipStream_t stream) {
    const float* logits = (const float*)d_in[0];
    const float* factor = (const float*)d_in[1];
    const int*   target = (const int*)d_in[2];
    float* partials = (float*)d_ws;   // 512 floats = 2 KB scratch
    float* out = (float*)d_out;

    weighted_lse_rowsum<<<kN / 64, 256, 0, stream>>>(logits, factor, target, partials);
    weighted_lse_finalize<<<1, 512, 0, stream>>>(partials, out);
}